// GGNLinearLayer_67276367724844
// MI455X (gfx1250) — compile-verified
//
#include <hip/hip_runtime.h>
#include <hip/hip_bf16.h>

// Problem constants (fixed by the reference).
#define Bn   4
#define Vn   8
#define Gn   65536      // H*W = 256*256
#define HWn  65536
#define Cin  32
#define Cout 32
#define Wdim 256
#define Hdim 256
#define STACKED_ELEMS ((size_t)Bn * Vn * Gn * Cout)   // 67,108,864

typedef __attribute__((ext_vector_type(2))) float v2f;
typedef __attribute__((ext_vector_type(4))) float v4f;
typedef __attribute__((ext_vector_type(8))) float v8f;

// ---------------------------------------------------------------------------
// Kernel 1: adjacency. One wave (32 threads), thread t = (b, row i).
// centers[b][v] = -R^T t ; sim = -dist^2 (diag -inf); top-3 mask;
// A = A_overlap*mask, diag=1; A_tilde = A / deg. A -> d_out tail, A_tilde -> ws.
// ---------------------------------------------------------------------------
__global__ __launch_bounds__(32) void adj_kernel(const float* __restrict__ extr,
                                                 float* __restrict__ A_out,
                                                 float* __restrict__ At_ws) {
    __shared__ float ctr[Bn * Vn][3];
    const int t = threadIdx.x;            // 0..31
    const int b = t >> 3;
    const int i = t & 7;

    // camera center for (b, i): c_k = -sum_j R[j][k] * trans[j], R[j][k] = E[j*4+k]
    const float* E = extr + (size_t)t * 16;
    #pragma unroll
    for (int k = 0; k < 3; ++k) {
        float s = 0.f;
        #pragma unroll
        for (int j = 0; j < 3; ++j) s += E[j * 4 + k] * E[j * 4 + 3];
        ctr[t][k] = -s;
    }
    __syncthreads();

    float sim[Vn], Aov[Vn];
    #pragma unroll
    for (int j = 0; j < Vn; ++j) {
        float dx = ctr[b * Vn + i][0] - ctr[b * Vn + j][0];
        float dy = ctr[b * Vn + i][1] - ctr[b * Vn + j][1];
        float dz = ctr[b * Vn + i][2] - ctr[b * Vn + j][2];
        float d2 = dx * dx + dy * dy + dz * dz;
        sim[j] = (j == i) ? -__builtin_huge_valf() : -d2;
        Aov[j] = 1.f / (1.f + sqrtf(d2 + 1e-6f));
    }
    // top-3 by sim (strict > => first index wins ties, matching lax.top_k)
    bool mask[Vn];
    #pragma unroll
    for (int j = 0; j < Vn; ++j) mask[j] = false;
    for (int kk = 0; kk < 3; ++kk) {
        int best = -1; float bv = -__builtin_huge_valf();
        for (int j = 0; j < Vn; ++j)
            if (!mask[j] && sim[j] > bv) { bv = sim[j]; best = j; }
        if (best >= 0) mask[best] = true;
    }
    float Arow[Vn], deg = 0.f;
    #pragma unroll
    for (int j = 0; j < Vn; ++j) {
        Arow[j] = (j == i) ? 1.f : (mask[j] ? Aov[j] : 0.f);
        deg += Arow[j];
    }
    const float dd = deg + ((deg == 0.f) ? 1.f : 0.f);
    #pragma unroll
    for (int j = 0; j < Vn; ++j) {
        A_out[(size_t)(b * Vn + i) * Vn + j] = Arow[j];
        At_ws[(size_t)(b * Vn + i) * Vn + j] = Arow[j] / dd;
    }
}

// ---------------------------------------------------------------------------
// Kernel 2: fused linear (WMMA f32 16x16x4, K chained 8x) + weighted scatter.
// One wave per 16 consecutive points of one (b, j) slab. EXEC is full
// (no divergent control flow) as WMMA requires.
//
// A-frag layout (ISA 7.12.2, 32-bit A 16x4): lane -> M = lane&15,
//   VGPR {0,1} -> K = 2*(lane>=16) + {0,1}.
// B-frag (4x16): lane -> N = lane&15, VGPR {0,1} -> K = 2*(lane>=16) + {0,1}.
// C/D (16x16 f32): VGPR r -> M = r + 8*(lane>=16), N = lane&15.
// ---------------------------------------------------------------------------
__global__ __launch_bounds__(256) void scatter_kernel(
    const float* __restrict__ feat,   // (B,V,G,Cin)
    const float* __restrict__ xy,     // (B,V,V,G,2)
    const float* __restrict__ Wl,     // (Cin,Cout)
    const float* __restrict__ bl,     // (Cout)
    const float* __restrict__ At,     // (B,V,V) A_tilde
    float* __restrict__ grids,        // (ni,B,HW,Cout)
    int i0, int ni)
{
    const int lane = threadIdx.x & 31;
    const int hl   = lane >> 4;                // which K-pair / M-half
    const int ln   = lane & 15;
    const int wave = blockIdx.x * (blockDim.x >> 5) + (threadIdx.x >> 5);

    const int bj = wave >> 12;                 // Gn/16 = 4096 tiles per (b,j)
    const int g0 = (wave & 4095) << 4;
    const int b  = bj >> 3;
    const int j  = bj & 7;

    // W fragments: constant across tiles; 2 column-halves x 8 K-chunks.
    v2f bf[16];
    #pragma unroll
    for (int h = 0; h < 2; ++h) {
        #pragma unroll
        for (int kc = 0; kc < 8; ++kc) {
            const int k = kc * 4 + hl * 2;
            const int n = h * 16 + ln;
            v2f w; w.x = Wl[k * Cout + n]; w.y = Wl[(k + 1) * Cout + n];
            bf[h * 8 + kc] = w;
        }
    }
    const float bias0 = bl[ln];
    const float bias1 = bl[16 + ln];

    // A fragments: 16 rows (points g0..g0+15) x Cin, streamed (non-temporal).
    const float* arow = feat + ((size_t)(b * Vn + j) * Gn + g0) * Cin;
    v2f af[8];
    #pragma unroll
    for (int kc = 0; kc < 8; ++kc) {
        const float* p = arow + (size_t)ln * Cin + kc * 4 + hl * 2;
        af[kc] = __builtin_nontemporal_load((const v2f*)p);
    }

    v8f c0, c1;
    #pragma unroll
    for (int r = 0; r < 8; ++r) { c0[r] = bias0; c1[r] = bias1; }

    #pragma unroll
    for (int kc = 0; kc < 8; ++kc) {
        c0 = __builtin_amdgcn_wmma_f32_16x16x4_f32(
                false, af[kc], false, bf[kc],     (short)0, c0, false, false);
        c1 = __builtin_amdgcn_wmma_f32_16x16x4_f32(
                false, af[kc], false, bf[8 + kc], (short)0, c1, false, false);
    }

    // Weighted scatter-add into per-target-view grids.
    const size_t xybase = (size_t)(b * Vn + j) * Vn * Gn * 2;
    #pragma unroll 1
    for (int ii = 0; ii < ni; ++ii) {
        const int i = i0 + ii;
        const float w = At[(size_t)(b * Vn + i) * Vn + j];
        float* gptr = grids + ((size_t)ii * Bn + b) * HWn * Cout;
        #pragma unroll
        for (int r = 0; r < 8; ++r) {
            const int g = g0 + r + hl * 8;     // C-frag row -> point index
            const v2f xyv = *(const v2f*)(xy + xybase + ((size_t)i * Gn + g) * 2);
            // rintf under default RNE mode == jnp.round (half-to-even)
            const int xi = (int)fminf(fmaxf(rintf(xyv.x), 0.f), (float)(Wdim - 1));
            const int yi = (int)fminf(fmaxf(rintf(xyv.y), 0.f), (float)(Hdim - 1));
            const int pix = yi * Wdim + xi;
            atomicAdd(gptr + (size_t)pix * Cout + ln,      c0[r] * w);
            atomicAdd(gptr + (size_t)pix * Cout + 16 + ln, c1[r] * w);
        }
    }
}

// ---------------------------------------------------------------------------
// Kernel 3: gather. out[b,i,g,:] = grid[ii][b][ idx[b,i,i,g] ][:], 128-bit lanes.
// ---------------------------------------------------------------------------
__global__ __launch_bounds__(256) void gather_kernel(
    const float* __restrict__ xy,
    const float* __restrict__ grids,
    float* __restrict__ out,
    int i0, int ni)
{
    const size_t t = (size_t)blockIdx.x * 256 + threadIdx.x;  // B*ni*G*8 threads
    const size_t p = t >> 3;
    const int    q = (int)(t & 7);
    const size_t g  = p & (Gn - 1);
    const size_t bi = p >> 16;
    const int ii = (int)(bi % (size_t)ni);
    const int b  = (int)(bi / (size_t)ni);
    const int i  = i0 + ii;

    const v2f xyv = *(const v2f*)(xy + (((size_t)(b * Vn + i) * Vn + i) * Gn + g) * 2);
    const int xi = (int)fminf(fmaxf(rintf(xyv.x), 0.f), (float)(Wdim - 1));
    const int yi = (int)fminf(fmaxf(rintf(xyv.y), 0.f), (float)(Hdim - 1));
    const int pix = yi * Wdim + xi;

    const v4f v = *(const v4f*)(grids +
        (((size_t)ii * Bn + b) * HWn + (size_t)pix) * Cout + (size_t)q * 4);
    __builtin_nontemporal_store(v,
        (v4f*)(out + (((size_t)(b * Vn + i) * Gn + g) * Cout) + (size_t)q * 4));
}

// ---------------------------------------------------------------------------
extern "C" void kernel_launch(void* const* d_in, const int* in_sizes, int n_in,
                              void* d_out, int out_size, void* d_ws, size_t ws_size,
                              hipStream_t stream) {
    const float* features = (const float*)d_in[0];
    const float* xy       = (const float*)d_in[1];
    const float* extr     = (const float*)d_in[2];
    const float* Wl       = (const float*)d_in[3];
    const float* bl       = (const float*)d_in[4];
    float* out = (float*)d_out;

    float* At    = (float*)d_ws;                       // 256 floats
    float* grids = (float*)((char*)d_ws + 4096);       // chunked per-i grids
    const size_t grid_bytes_per_i = (size_t)Bn * HWn * Cout * sizeof(float); // 32 MB

    int chunk = 1;
    if (ws_size > 4096) {
        size_t fit = (ws_size - 4096) / grid_bytes_per_i;
        chunk = (int)(fit < 1 ? 1 : (fit > Vn ? Vn : fit));
    }

    adj_kernel<<<1, 32, 0, stream>>>(extr, out + STACKED_ELEMS, At);

    for (int i0 = 0; i0 < Vn; i0 += chunk) {
        const int ni = (i0 + chunk <= Vn) ? chunk : (Vn - i0);
        (void)hipMemsetAsync(grids, 0, grid_bytes_per_i * (size_t)ni, stream);
        // waves = B*V*G/16 = 131072; 8 waves per 256-thread block
        scatter_kernel<<<16384, 256, 0, stream>>>(features, xy, Wl, bl, At,
                                                  grids, i0, ni);
        gather_kernel<<<(unsigned)((size_t)Bn * ni * Gn * 8 / 256), 256, 0, stream>>>(
            xy, grids, out, i0, ni);
    }
}